// DIT_11458972746143
// MI455X (gfx1250) — compile-verified
//
#include <hip/hip_runtime.h>
#include <math.h>

#define BATCH 16
#define NPTS  2048
#define KNN   10
#define SHARPF 30.0f
#define THRESHF 0.6f
#define EPSF 1e-6f

typedef __attribute__((ext_vector_type(2))) float v2f;
typedef __attribute__((ext_vector_type(8))) float v8f;

// ---------------------------------------------------------------------------
// Kernel 1: per (batch, 16-row tile): Gram matrix via V_WMMA_F32_16X16X4_F32,
// d2 = |xi|^2 + |xj|^2 - 2*gram, mask (d2+1e-7 < 0.1 -> excluded), streaming
// stable top-10 (ascending d2, earlier index wins ties) per row.
// grid = BATCH * (NPTS/16) blocks, 256 threads = 8 waves.
// ---------------------------------------------------------------------------
__global__ __launch_bounds__(256) void knn_wmma_kernel(const float* __restrict__ src,
                                                       int* __restrict__ knn_idx) {
  __shared__ float pts[NPTS][4];   // x,y,z,0   (32 KB)
  __shared__ float sqv[NPTS];      // |p|^2     (8 KB)
  __shared__ float d2t[16][260];   // 256-col d2 chunk (+pad)  (~16.6 KB)

  const int tid = threadIdx.x;
  const int b   = blockIdx.x / (NPTS / 16);
  const int m0  = (blockIdx.x % (NPTS / 16)) * 16;
  const float* sb = src + (size_t)b * 3 * NPTS;

  // Phase 0: stage this batch's points (padded to K=4) + squared norms.
  for (int p = tid; p < NPTS; p += 256) {
    float x = sb[0 * NPTS + p];
    float y = sb[1 * NPTS + p];
    float z = sb[2 * NPTS + p];
    pts[p][0] = x; pts[p][1] = y; pts[p][2] = z; pts[p][3] = 0.0f;
    sqv[p] = x * x + y * y + z * z;
  }
  __syncthreads();

  const int lane = tid & 31;
  const int wave = tid >> 5;
  const int half = lane >> 4;   // K-pair select: {0,1} vs {2,3}
  const int l16  = lane & 15;

  // A matrix (16x4 f32): lane l16 = row m0+l16; VGPR0=K(2*half), VGPR1=K(2*half+1)
  v2f amat;
  amat.x = pts[m0 + l16][2 * half + 0];
  amat.y = pts[m0 + l16][2 * half + 1];

  float bd[KNN]; int bi[KNN];
#pragma unroll
  for (int k = 0; k < KNN; ++k) { bd[k] = 3.0e38f; bi[k] = 0; }

  for (int chunk = 0; chunk < NPTS / 256; ++chunk) {
    // Each wave produces two 16x16 tiles -> 8 waves cover 256 columns.
#pragma unroll
    for (int t = 0; t < 2; ++t) {
      const int nbase = chunk * 256 + wave * 32 + t * 16;
      v2f bmat;                       // B matrix (4x16): same striping, N=l16
      bmat.x = pts[nbase + l16][2 * half + 0];
      bmat.y = pts[nbase + l16][2 * half + 1];
      v8f acc = {};
      acc = __builtin_amdgcn_wmma_f32_16x16x4_f32(false, amat, false, bmat,
                                                  (short)0, acc, false, false);
#pragma unroll
      for (int v = 0; v < 8; ++v) {
        const int ml = v + 8 * half;            // C/D layout: M = v + 8*(lane/16)
        const int nl = wave * 32 + t * 16 + l16;
        const int ng = chunk * 256 + nl;
        float d2 = sqv[m0 + ml] + sqv[ng] - 2.0f * acc[v];
        // reference: distance=-(d2+1e-7), masked to -inf when distance > -0.1
        if (d2 + 1e-7f < 0.1f) d2 = 3.0e38f;
        d2t[ml][nl] = d2;
      }
    }
    __syncthreads();

    // Stable streaming top-10 per row (threads 0..15, ascending index order).
    if (tid < 16) {
      for (int j = 0; j < 256; ++j) {
        float vj = d2t[tid][j];
        if (vj < bd[KNN - 1]) {
          bd[KNN - 1] = vj; bi[KNN - 1] = chunk * 256 + j;
#pragma unroll
          for (int q = KNN - 1; q > 0; --q) {
            if (bd[q] < bd[q - 1]) {          // strict < keeps earlier index first on ties
              float tf = bd[q]; bd[q] = bd[q - 1]; bd[q - 1] = tf;
              int   ti = bi[q]; bi[q] = bi[q - 1]; bi[q - 1] = ti;
            }
          }
        }
      }
    }
    __syncthreads();
  }

  if (tid < 16) {
    int* o = knn_idx + ((size_t)b * NPTS + m0 + tid) * KNN;
#pragma unroll
    for (int k = 0; k < KNN; ++k) o[k] = bi[k];
  }
}

// ---------------------------------------------------------------------------
// Kernel 2: per point: 45 triangles (self + neighbor pair), sorted sq side
// lengths src vs tgt(+eps), relative loss, keep 10 smallest, mean sqrt.
// ---------------------------------------------------------------------------
__device__ __forceinline__ void sort3(float& a, float& b, float& c) {
  float t;
  if (a > b) { t = a; a = b; b = t; }
  if (b > c) { t = b; b = c; c = t; }
  if (a > b) { t = a; a = b; b = t; }
}
__device__ __forceinline__ float sq3(float dx, float dy, float dz) {
  return dx * dx + dy * dy + dz * dz;
}

__global__ __launch_bounds__(256) void tri_loss_kernel(const float* __restrict__ src,
                                                       const float* __restrict__ tgt,
                                                       const int* __restrict__ knn_idx,
                                                       float* __restrict__ loss) {
  int g = blockIdx.x * 256 + threadIdx.x;
  if (g >= BATCH * NPTS) return;
  int b = g / NPTS, n = g % NPTS;
  const float* sb = src + (size_t)b * 3 * NPTS;
  const float* tb = tgt + (size_t)b * 3 * NPTS;

  float sx[11], sy[11], sz[11], tx[11], ty[11], tz[11];
  sx[0] = sb[n]; sy[0] = sb[NPTS + n]; sz[0] = sb[2 * NPTS + n];
  tx[0] = tb[n]; ty[0] = tb[NPTS + n]; tz[0] = tb[2 * NPTS + n];
  const int* nb = knn_idx + (size_t)g * KNN;
#pragma unroll
  for (int i = 0; i < KNN; ++i) {
    int j = nb[i];
    sx[i + 1] = sb[j]; sy[i + 1] = sb[NPTS + j]; sz[i + 1] = sb[2 * NPTS + j];
    tx[i + 1] = tb[j]; ty[i + 1] = tb[NPTS + j]; tz[i + 1] = tb[2 * NPTS + j];
  }

  float best[KNN];
#pragma unroll
  for (int k = 0; k < KNN; ++k) best[k] = 3.0e38f;

#pragma unroll
  for (int i = 0; i < KNN; ++i) {
#pragma unroll
    for (int j = i + 1; j < KNN; ++j) {
      // triangle (self, n_i, n_j): d01, d12, d02 -> sorted ascending
      float a0 = sq3(sx[0] - sx[i + 1], sy[0] - sy[i + 1], sz[0] - sz[i + 1]);
      float a1 = sq3(sx[i + 1] - sx[j + 1], sy[i + 1] - sy[j + 1], sz[i + 1] - sz[j + 1]);
      float a2 = sq3(sx[0] - sx[j + 1], sy[0] - sy[j + 1], sz[0] - sz[j + 1]);
      sort3(a0, a1, a2);
      float c0 = sq3(tx[0] - tx[i + 1], ty[0] - ty[i + 1], tz[0] - tz[i + 1]);
      float c1 = sq3(tx[i + 1] - tx[j + 1], ty[i + 1] - ty[j + 1], tz[i + 1] - tz[j + 1]);
      float c2 = sq3(tx[0] - tx[j + 1], ty[0] - ty[j + 1], tz[0] - tz[j + 1]);
      sort3(c0, c1, c2);
      c0 += EPSF; c1 += EPSF; c2 += EPSF;
      float d0 = a0 - c0, d1 = a1 - c1, d2 = a2 - c2;
      float s0 = a0 + c0, s1 = a1 + c1, s2 = a2 + c2;
      float v = (d0 * d0 + d1 * d1 + d2 * d2) / (s0 * s0 + s1 * s1 + s2 * s2);
      // keep 10 smallest
      if (v < best[KNN - 1]) {
        best[KNN - 1] = v;
#pragma unroll
        for (int q = KNN - 1; q > 0; --q) {
          if (best[q] < best[q - 1]) { float tf = best[q]; best[q] = best[q - 1]; best[q - 1] = tf; }
        }
      }
    }
  }

  float acc = 0.0f;
#pragma unroll
  for (int k = 0; k < KNN; ++k) acc += sqrtf(best[k] + EPSF);
  loss[g] = acc * (1.0f / KNN);
}

// ---------------------------------------------------------------------------
// Kernel 3: per batch: min over N, weight = 2*sigmoid(-30*(loss-min)) > 0.6
// ---------------------------------------------------------------------------
__global__ __launch_bounds__(256) void finalize_kernel(const float* __restrict__ loss,
                                                       float* __restrict__ out) {
  __shared__ float red[256];
  const int b = blockIdx.x, tid = threadIdx.x;
  const float* lb = loss + (size_t)b * NPTS;
  float mn = 3.0e38f;
  for (int k = tid; k < NPTS; k += 256) mn = fminf(mn, lb[k]);
  red[tid] = mn;
  __syncthreads();
  for (int s = 128; s > 0; s >>= 1) {
    if (tid < s) red[tid] = fminf(red[tid], red[tid + s]);
    __syncthreads();
  }
  float mv = red[0];
  for (int k = tid; k < NPTS; k += 256) {
    float x = lb[k] - mv;
    float w = 2.0f / (1.0f + __expf(SHARPF * x));   // 2*sigmoid(-SHARP*x)
    out[(size_t)b * NPTS + k] = (w > THRESHF) ? 1.0f : 0.0f;
  }
}

// ---------------------------------------------------------------------------
extern "C" void kernel_launch(void* const* d_in, const int* in_sizes, int n_in,
                              void* d_out, int out_size, void* d_ws, size_t ws_size,
                              hipStream_t stream) {
  const float* src = (const float*)d_in[0];
  const float* tgt = (const float*)d_in[1];
  float* out = (float*)d_out;

  // workspace layout: [knn indices: B*N*K ints][loss: B*N floats]
  int* knn_idx = (int*)d_ws;
  float* loss = (float*)((char*)d_ws + (size_t)BATCH * NPTS * KNN * sizeof(int));

  knn_wmma_kernel<<<BATCH * (NPTS / 16), 256, 0, stream>>>(src, knn_idx);
  tri_loss_kernel<<<(BATCH * NPTS + 255) / 256, 256, 0, stream>>>(src, tgt, knn_idx, loss);
  finalize_kernel<<<BATCH, 256, 0, stream>>>(loss, out);
}